// SparseMoE_21234318311690
// MI455X (gfx1250) — compile-verified
//
#include <hip/hip_runtime.h>
#include <stdint.h>

typedef float v2f __attribute__((ext_vector_type(2)));
typedef float v8f __attribute__((ext_vector_type(8)));

#define T_TOK 2048
#define Dm 1024
#define Em 8
#define Fm 4096
#define TILE_M 32
#define XS_STRIDE (Dm + 4)    // pad to dodge LDS bank conflicts (64 banks)
#define HS_STRIDE (128 + 4)
#define GTHRESH 1e-9f

// ---------------- zero out + counters ----------------
__global__ void zero_kernel(float* __restrict__ out, int n, int* __restrict__ cnt) {
    int i = blockIdx.x * blockDim.x + threadIdx.x;
    if (i < n) out[i] = 0.0f;
    if (i < Em) cnt[i] = 0;
}

// ---------------- router logits + alpha-entmax (alpha=1.5) + expert lists ----------------
__global__ __launch_bounds__(256) void router_kernel(
    const float* __restrict__ x, const float* __restrict__ Wr, const float* __restrict__ br,
    int* __restrict__ cnt, int* __restrict__ lists, float* __restrict__ gates)
{
    __shared__ float red[Em * 256];
    const int t = blockIdx.x;
    const int tid = threadIdx.x;

    float acc[Em];
#pragma unroll
    for (int e = 0; e < Em; ++e) acc[e] = 0.0f;

    const float* xr = x + (size_t)t * Dm;
    for (int d = tid; d < Dm; d += 256) {
        float xv = xr[d];
        const float* wr = Wr + (size_t)d * Em;
#pragma unroll
        for (int e = 0; e < Em; ++e) acc[e] += xv * wr[e];
    }
#pragma unroll
    for (int e = 0; e < Em; ++e) red[e * 256 + tid] = acc[e];
    __syncthreads();
    for (int off = 128; off > 0; off >>= 1) {
        if (tid < off) {
#pragma unroll
            for (int e = 0; e < Em; ++e) red[e * 256 + tid] += red[e * 256 + tid + off];
        }
        __syncthreads();
    }

    if (tid == 0) {
        // Xs = logits * (alpha-1), TEMP=1, alpha-1 = 0.5, inv_am1 = 2 -> square
        float Xs[Em];
        float mx = -3.0e38f;
#pragma unroll
        for (int e = 0; e < Em; ++e) {
            Xs[e] = 0.5f * (red[e * 256] + br[e]);
            mx = fmaxf(mx, Xs[e]);
        }
        float tau_lo = mx - 1.0f;
        float tau_hi = mx - 0.35355339059327373f;   // (1/8)^(alpha-1)
        float f_lo = -1.0f;
#pragma unroll
        for (int e = 0; e < Em; ++e) { float v = fmaxf(Xs[e] - tau_lo, 0.0f); f_lo += v * v; }
        float dm = tau_hi - tau_lo;
        float tau_m = tau_lo;
        for (int it = 0; it < 25; ++it) {
            dm *= 0.5f;
            tau_m = tau_lo + dm;
            float f_m = -1.0f;
#pragma unroll
            for (int e = 0; e < Em; ++e) { float v = fmaxf(Xs[e] - tau_m, 0.0f); f_m += v * v; }
            if (f_m * f_lo >= 0.0f) tau_lo = tau_m;
        }
        float p[Em]; float s = 0.0f;
#pragma unroll
        for (int e = 0; e < Em; ++e) { float v = fmaxf(Xs[e] - tau_m, 0.0f); p[e] = v * v; s += p[e]; }
        float inv = 1.0f / s;
#pragma unroll
        for (int e = 0; e < Em; ++e) {
            float g = p[e] * inv;
            g = (g > GTHRESH) ? g : 0.0f;
            gates[t * Em + e] = g;
            if (g > 0.0f) {
                int i = atomicAdd(&cnt[e], 1);
                lists[e * T_TOK + i] = t;
            }
        }
    }
}

// ---------------- fused expert FFN over compacted 32-token tiles (f32 WMMA) ----------------
__global__ __launch_bounds__(256) void moe_ffn_kernel(
    const float* __restrict__ x,
    const float* __restrict__ W1, const float* __restrict__ b1,
    const float* __restrict__ W2, const float* __restrict__ b2,
    const int* __restrict__ cnt, const int* __restrict__ lists,
    const float* __restrict__ gates, float* __restrict__ out)
{
    extern __shared__ float smem[];
    float* xs = smem;                            // [32][XS_STRIDE]
    float* hs = smem + TILE_M * XS_STRIDE;       // [32][HS_STRIDE]
    __shared__ int   toks[TILE_M];
    __shared__ float gs[TILE_M];

    const int e = blockIdx.y;
    const int tile = blockIdx.x;
    const int cnt_e = cnt[e];
    if (tile * TILE_M >= cnt_e) return;          // inactive tile: whole block exits

    const int tid  = threadIdx.x;
    const int lane = tid & 31;
    const int wave = tid >> 5;                   // 0..7
    const int half = lane >> 4;                  // 0/1 : K-split half of the wave
    const int l16  = lane & 15;

    const float* W1e = W1 + (size_t)e * Dm * Fm;
    const float* b1e = b1 + (size_t)e * Fm;
    const float* W2e = W2 + (size_t)e * Fm * Dm;
    const float* b2e = b2 + (size_t)e * Dm;

    if (tid < TILE_M) {
        int idx = tile * TILE_M + tid;
        int tok = (idx < cnt_e) ? lists[e * T_TOK + idx] : -1;
        toks[tid] = tok;
        gs[tid] = (tok >= 0) ? gates[tok * Em + e] : 0.0f;
    }
    __syncthreads();

    // ---- async gather of x tile into LDS (global_load_async_to_lds_b128, ASYNCcnt) ----
    // 32 rows x 256 float4; 8 threads per row, 32 b128 transfers per thread.
    // Padded rows (tok < 0) are not fetched: WMMA row M only depends on A row M, and
    // their scatter is skipped, so stale LDS there is harmless.
    {
        const int r = tid >> 3;
        const int c = tid & 7;
        int tok = toks[r];
        if (tok >= 0) {
            const float* src = x + (size_t)tok * Dm;
            uint32_t ldsrow = (uint32_t)(uintptr_t)(xs + r * XS_STRIDE);
#pragma unroll
            for (int i = 0; i < 32; ++i) {
                int c4 = c + i * 8;                              // float4 index in row
                uint64_t ga = (uint64_t)(uintptr_t)(src + c4 * 4);
                uint32_t la = ldsrow + (uint32_t)(c4 * 16);
                asm volatile("global_load_async_to_lds_b128 %0, %1, off"
                             :: "v"(la), "v"(ga) : "memory");
            }
        }
    }
    asm volatile("s_wait_asynccnt 0x0" ::: "memory");
    __syncthreads();

    const v8f z8 = {0.f, 0.f, 0.f, 0.f, 0.f, 0.f, 0.f, 0.f};
    v8f yacc0[8], yacc1[8];
#pragma unroll
    for (int nb = 0; nb < 8; ++nb) { yacc0[nb] = z8; yacc1[nb] = z8; }

    // A-frag bases (16x4 f32 layout: lanes 0-15 -> K0,K1; lanes 16-31 -> K2,K3)
    const float* aB0 = xs + l16 * XS_STRIDE + 2 * half;           // tile rows 0..15
    const float* aB1 = xs + (l16 + 16) * XS_STRIDE + 2 * half;    // tile rows 16..31

    for (int fc = 0; fc < Fm; fc += 128) {
        // ---- matmul1: h[32][fc..fc+128) = relu(xs @ W1 + b1); each wave: one 16-wide
        //      column block, two 16-row halves sharing every B fragment.
        v8f hacc0 = z8, hacc1 = z8;
        const float* w1p = W1e + (size_t)(2 * half) * Fm + (fc + wave * 16 + l16);
        for (int k = 0; k < Dm; k += 4) {
            v2f a0 = *(const v2f*)(aB0 + k);
            v2f a1 = *(const v2f*)(aB1 + k);
            v2f b;
            b.x = w1p[(size_t)k * Fm];
            b.y = w1p[(size_t)(k + 1) * Fm];
            hacc0 = __builtin_amdgcn_wmma_f32_16x16x4_f32(false, a0, false, b,
                                                          (short)0, hacc0, false, false);
            hacc1 = __builtin_amdgcn_wmma_f32_16x16x4_f32(false, a1, false, b,
                                                          (short)0, hacc1, false, false);
        }
        float b1v = b1e[fc + wave * 16 + l16];
#pragma unroll
        for (int j = 0; j < 8; ++j) {        // C layout: M = j + 8*half, N = l16
            float v0 = hacc0[j] + b1v;
            float v1 = hacc1[j] + b1v;
            hs[(j + 8 * half) * HS_STRIDE + wave * 16 + l16]        = fmaxf(v0, 0.0f);
            hs[(16 + j + 8 * half) * HS_STRIDE + wave * 16 + l16]   = fmaxf(v1, 0.0f);
        }
        __syncthreads();

        // prefetch next chunk of W2 into cache (gfx1250 global_prefetch_b8)
        if (fc + 128 < Fm)
            __builtin_prefetch(W2e + (size_t)(fc + 128 + 2 * half) * Dm + wave * 128 + l16, 0, 1);

        // ---- matmul2: y[32][wave*128 .. +128) += hs @ W2[fc:fc+128, :]
        //      each B fragment feeds two WMMAs (row halves) -> half the loads per FLOP.
        const float* hA0 = hs + l16 * HS_STRIDE + 2 * half;
        const float* hA1 = hs + (l16 + 16) * HS_STRIDE + 2 * half;
        const float* w2p = W2e + (size_t)(fc + 2 * half) * Dm + (wave * 128 + l16);
        for (int k = 0; k < 128; k += 4) {
            v2f a0 = *(const v2f*)(hA0 + k);
            v2f a1 = *(const v2f*)(hA1 + k);
            const float* wk = w2p + (size_t)k * Dm;
#pragma unroll
            for (int nb = 0; nb < 8; ++nb) {
                v2f b;
                b.x = wk[nb * 16];
                b.y = wk[nb * 16 + Dm];
                yacc0[nb] = __builtin_amdgcn_wmma_f32_16x16x4_f32(false, a0, false, b,
                                                                  (short)0, yacc0[nb], false, false);
                yacc1[nb] = __builtin_amdgcn_wmma_f32_16x16x4_f32(false, a1, false, b,
                                                                  (short)0, yacc1[nb], false, false);
            }
        }
        __syncthreads();                     // hs is overwritten next chunk
    }

    // ---- epilogue: out[tok] += g * (y + b2), atomic scatter ----
    int   tk0[8], tk1[8];
    float g0[8],  g1[8];
#pragma unroll
    for (int j = 0; j < 8; ++j) {
        int m = j + 8 * half;
        tk0[j] = toks[m];      g0[j] = gs[m];
        tk1[j] = toks[16 + m]; g1[j] = gs[16 + m];
    }
#pragma unroll
    for (int nb = 0; nb < 8; ++nb) {
        int col = wave * 128 + nb * 16 + l16;
        float b2v = b2e[col];
#pragma unroll
        for (int j = 0; j < 8; ++j) {
            if (tk0[j] >= 0 && g0[j] > 0.0f)
                atomicAdd(out + (size_t)tk0[j] * Dm + col, g0[j] * (yacc0[nb][j] + b2v));
            if (tk1[j] >= 0 && g1[j] > 0.0f)
                atomicAdd(out + (size_t)tk1[j] * Dm + col, g1[j] * (yacc1[nb][j] + b2v));
        }
    }
}

extern "C" void kernel_launch(void* const* d_in, const int* in_sizes, int n_in,
                              void* d_out, int out_size, void* d_ws, size_t ws_size,
                              hipStream_t stream) {
    (void)in_sizes; (void)n_in; (void)ws_size; (void)out_size;
    const float* x  = (const float*)d_in[0];
    const float* Wr = (const float*)d_in[1];
    const float* br = (const float*)d_in[2];
    const float* W1 = (const float*)d_in[3];
    const float* b1 = (const float*)d_in[4];
    const float* W2 = (const float*)d_in[5];
    const float* b2 = (const float*)d_in[6];
    float* out = (float*)d_out;

    // workspace layout
    int* cnt     = (int*)d_ws;                       // [E]
    int* lists   = cnt + Em;                         // [E][T]
    float* gates = (float*)(lists + Em * T_TOK);     // [T][E]

    const int nOut = T_TOK * Dm;
    zero_kernel<<<(nOut + 255) / 256, 256, 0, stream>>>(out, nOut, cnt);
    router_kernel<<<T_TOK, 256, 0, stream>>>(x, Wr, br, cnt, lists, gates);

    size_t shmem = (size_t)(TILE_M * XS_STRIDE + TILE_M * HS_STRIDE) * sizeof(float); // ~145 KB (< 320 KB WGP LDS)
    dim3 grid(T_TOK / TILE_M, Em);
    moe_ffn_kernel<<<grid, 256, shmem, stream>>>(x, W1, b1, W2, b2, cnt, lists, gates, out);
}